// Encode_Graph_73976516706557
// MI455X (gfx1250) — compile-verified
//
#include <hip/hip_runtime.h>

#define N_NODES  500000
#define N_EDGES  1000000
#define N_GRAPHS 512
#define N_TILES  (N_NODES / 16)   // 31250 exactly, no tail

typedef __attribute__((ext_vector_type(2))) float v2f;
typedef __attribute__((ext_vector_type(8))) float v8f;

// ---------------------------------------------------------------------------
// Wave-level f32 GEMM building block: C[16xN] += A[16xK] * W[KxN]
// using V_WMMA_F32_16X16X4_F32 chained over K in steps of 4.
// A fragment layout (ISA 7.12.2, 32-bit A 16x4): lane l holds
//   A[l&15][kb], A[l&15][kb+1] with kb = 4*kk + 2*(l>>4).
// B fragment mirrors A with N in place of M.
// C/D layout: vgpr r, lane l -> row = r + 8*(l>>4), col = l&15 (per N-tile).
// ---------------------------------------------------------------------------
template <int K, int N>
__device__ inline void wave_gemm_acc(v8f* acc, const float* __restrict__ A,
                                     int lda, const float* __restrict__ sW,
                                     int lane, float ascale) {
  const int r16 = lane & 15;
  const int hs2 = (lane >> 4) << 1;
#pragma unroll
  for (int kk = 0; kk < K / 4; ++kk) {
    const int kb = kk * 4 + hs2;
    v2f a;
    a.x = A[r16 * lda + kb] * ascale;
    a.y = A[r16 * lda + kb + 1] * ascale;
#pragma unroll
    for (int nt = 0; nt < N / 16; ++nt) {
      const int col = nt * 16 + r16;
      v2f b;
      b.x = sW[kb * N + col];
      b.y = sW[(kb + 1) * N + col];
      acc[nt] = __builtin_amdgcn_wmma_f32_16x16x4_f32(
          false, a, false, b, (short)0, acc[nt], false, false);
    }
  }
}

template <int N>
__device__ inline void store_bias_relu(float* __restrict__ D, int tile,
                                       const v8f* acc,
                                       const float* __restrict__ sB, int lane) {
  const int r16 = lane & 15;
  const int hs = lane >> 4;
  const size_t rowBase = (size_t)tile * 16 + hs * 8;
#pragma unroll
  for (int nt = 0; nt < N / 16; ++nt) {
    const int col = nt * 16 + r16;
    const float bias = sB[col];
#pragma unroll
    for (int r = 0; r < 8; ++r) {
      float v = acc[nt][r] + bias;
      D[(rowBase + r) * (size_t)N + col] = fmaxf(v, 0.0f);
    }
  }
}

// ---------------------------------------------------------------------------
// Stage 0: zeroing (d_out + accumulated ws regions), float4-wide
// ---------------------------------------------------------------------------
__global__ void zero4_kernel(float4* __restrict__ p, long n4) {
  long i = (long)blockIdx.x * blockDim.x + threadIdx.x;
  if (i < n4) p[i] = make_float4(0.f, 0.f, 0.f, 0.f);
}

// ---------------------------------------------------------------------------
// Stage 1: degree counts (atomic f32 add of 1.0 per edge endpoint)
// ---------------------------------------------------------------------------
__global__ void degree_kernel(const int* __restrict__ src,
                              const int* __restrict__ dst,
                              float* __restrict__ degO,
                              float* __restrict__ degI) {
  int e = blockIdx.x * blockDim.x + threadIdx.x;
  if (e < N_EDGES) {
    atomicAdd(&degO[src[e]], 1.0f);
    atomicAdd(&degI[dst[e]], 1.0f);
  }
}

// Stage 2: deg -> clip(deg,1)^-0.5 in place (both norms in one pass)
__global__ void norm_kernel(float* __restrict__ degO, float* __restrict__ degI) {
  int i = blockIdx.x * blockDim.x + threadIdx.x;
  if (i < N_NODES) {
    degO[i] = rsqrtf(fmaxf(degO[i], 1.0f));
    degI[i] = rsqrtf(fmaxf(degI[i], 1.0f));
  }
}

// ---------------------------------------------------------------------------
// Stage 3: x[i] = embed[tokens[i]] * norm_src[i]   (float4-wide, 4 thr/node)
// ---------------------------------------------------------------------------
__global__ void embed_scale_kernel(const int* __restrict__ tokens,
                                   const float* __restrict__ embed,
                                   const float* __restrict__ norm_src,
                                   float* __restrict__ x) {
  long t = (long)blockIdx.x * blockDim.x + threadIdx.x;
  if (t >= (long)N_NODES * 4) return;
  int i = (int)(t >> 2), q = (int)(t & 3);
  int tok = tokens[i];
  float ns = norm_src[i];
  float4 v = ((const float4*)embed)[(size_t)tok * 4 + q];
  v.x *= ns; v.y *= ns; v.z *= ns; v.w *= ns;
  ((float4*)x)[(size_t)i * 4 + q] = v;
}

// Stage 4: m[dst] += x[src]   (16 floats per edge, 4 threads/edge)
__global__ void scatter_add_kernel(const int* __restrict__ src,
                                   const int* __restrict__ dst,
                                   const float* __restrict__ x,
                                   float* __restrict__ m) {
  long t = (long)blockIdx.x * blockDim.x + threadIdx.x;
  if (t >= (long)N_EDGES * 4) return;
  int e = (int)(t >> 2), q = (int)(t & 3);
  int s = src[e], d = dst[e];
  float4 v = ((const float4*)x)[(size_t)s * 4 + q];
  float* o = m + (size_t)d * 16 + q * 4;
  atomicAdd(o + 0, v.x);
  atomicAdd(o + 1, v.y);
  atomicAdd(o + 2, v.z);
  atomicAdd(o + 3, v.w);
}

// ---------------------------------------------------------------------------
// Stage 5: h1 = relu((m * norm_dst) @ W1[16x32] + b1)     WMMA f32, K=16,N=32
// ---------------------------------------------------------------------------
__global__ void __launch_bounds__(128) gemm1_kernel(
    const float* __restrict__ m, const float* __restrict__ norm_dst,
    const float* __restrict__ W1, const float* __restrict__ b1,
    float* __restrict__ h1) {
  __shared__ float sW[16 * 32];
  __shared__ float sB[32];
  for (int i = threadIdx.x; i < 16 * 32; i += blockDim.x) sW[i] = W1[i];
  if (threadIdx.x < 32) sB[threadIdx.x] = b1[threadIdx.x];
  __syncthreads();
  int lane = threadIdx.x & 31;
  int tile = blockIdx.x * (blockDim.x >> 5) + (threadIdx.x >> 5);
  if (tile >= N_TILES) return;
  float nd = norm_dst[tile * 16 + (lane & 15)];
  v8f acc[2] = {};
  wave_gemm_acc<16, 32>(acc, m + (size_t)tile * 16 * 16, 16, sW, lane, nd);
  store_bias_relu<32>(h1, tile, acc, sB, lane);
}

// Stage 6: hp = relu(h1 @ Wpool[32x32] + bpool)           WMMA f32, K=32,N=32
__global__ void __launch_bounds__(128) gemm_pool_kernel(
    const float* __restrict__ h1, const float* __restrict__ Wpool,
    const float* __restrict__ bpool, float* __restrict__ hp) {
  __shared__ float sW[32 * 32];
  __shared__ float sB[32];
  for (int i = threadIdx.x; i < 32 * 32; i += blockDim.x) sW[i] = Wpool[i];
  if (threadIdx.x < 32) sB[threadIdx.x] = bpool[threadIdx.x];
  __syncthreads();
  int lane = threadIdx.x & 31;
  int tile = blockIdx.x * (blockDim.x >> 5) + (threadIdx.x >> 5);
  if (tile >= N_TILES) return;
  v8f acc[2] = {};
  wave_gemm_acc<32, 32>(acc, h1 + (size_t)tile * 16 * 32, 32, sW, lane, 1.0f);
  store_bias_relu<32>(hp, tile, acc, sB, lane);
}

// Stage 7: hn[dst] = max(hn[dst], hp[src]); hn pre-zeroed. hp >= 0 (relu), so
// integer atomicMax on the IEEE bit pattern is exact, and isolated nodes end
// at 0 exactly like DGL's isfinite->0 rule. 8 threads/edge.
__global__ void scatter_max_kernel(const int* __restrict__ src,
                                   const int* __restrict__ dst,
                                   const float* __restrict__ hp,
                                   float* __restrict__ hn) {
  long t = (long)blockIdx.x * blockDim.x + threadIdx.x;
  if (t >= (long)N_EDGES * 8) return;
  int e = (int)(t >> 3), q = (int)(t & 7);
  int s = src[e], d = dst[e];
  float4 v = ((const float4*)hp)[(size_t)s * 8 + q];
  int* o = (int*)(hn + (size_t)d * 32 + q * 4);
  atomicMax(o + 0, __float_as_int(v.x));
  atomicMax(o + 1, __float_as_int(v.y));
  atomicMax(o + 2, __float_as_int(v.z));
  atomicMax(o + 3, __float_as_int(v.w));
}

// Stage 8: h2 = relu(h1 @ Wself + hn @ Wneigh + bneigh)   2x WMMA K=32,N=64
__global__ void __launch_bounds__(128) gemm2_kernel(
    const float* __restrict__ h1, const float* __restrict__ hn,
    const float* __restrict__ Wself, const float* __restrict__ Wneigh,
    const float* __restrict__ bneigh, float* __restrict__ h2) {
  __shared__ float sWs[32 * 64];
  __shared__ float sWn[32 * 64];
  __shared__ float sB[64];
  for (int i = threadIdx.x; i < 32 * 64; i += blockDim.x) {
    sWs[i] = Wself[i];
    sWn[i] = Wneigh[i];
  }
  if (threadIdx.x < 64) sB[threadIdx.x] = bneigh[threadIdx.x];
  __syncthreads();
  int lane = threadIdx.x & 31;
  int tile = blockIdx.x * (blockDim.x >> 5) + (threadIdx.x >> 5);
  if (tile >= N_TILES) return;
  v8f acc[4] = {};
  wave_gemm_acc<32, 64>(acc, h1 + (size_t)tile * 16 * 32, 32, sWs, lane, 1.0f);
  wave_gemm_acc<32, 64>(acc, hn + (size_t)tile * 16 * 32, 32, sWn, lane, 1.0f);
  store_bias_relu<64>(h2, tile, acc, sB, lane);
}

// Stage 9: h3 = relu(h2 @ Wlin[64x64] + blin), fused segment-sum into
// d_out[graph_ids[node]]: run-length compress the (sorted) graph ids per
// 8-row lane strip before issuing f32 atomics. WMMA K=64, N=64.
__global__ void __launch_bounds__(128) gemm3_kernel(
    const float* __restrict__ h2, const float* __restrict__ Wlin,
    const float* __restrict__ blin, const int* __restrict__ graph_ids,
    float* __restrict__ out) {
  __shared__ float sW[64 * 64];
  __shared__ float sB[64];
  for (int i = threadIdx.x; i < 64 * 64; i += blockDim.x) sW[i] = Wlin[i];
  if (threadIdx.x < 64) sB[threadIdx.x] = blin[threadIdx.x];
  __syncthreads();
  int lane = threadIdx.x & 31;
  int tile = blockIdx.x * (blockDim.x >> 5) + (threadIdx.x >> 5);
  if (tile >= N_TILES) return;
  v8f acc[4] = {};
  wave_gemm_acc<64, 64>(acc, h2 + (size_t)tile * 16 * 64, 64, sW, lane, 1.0f);

  const int r16 = lane & 15, hs = lane >> 4;
  const int nodeBase = tile * 16 + hs * 8;
  int curGid = graph_ids[nodeBase];
  float sum[4] = {0.f, 0.f, 0.f, 0.f};
#pragma unroll
  for (int rr = 0; rr < 8; ++rr) {
    int gid = graph_ids[nodeBase + rr];
    if (gid != curGid) {
#pragma unroll
      for (int nt = 0; nt < 4; ++nt)
        atomicAdd(&out[(size_t)curGid * 64 + nt * 16 + r16], sum[nt]);
      sum[0] = sum[1] = sum[2] = sum[3] = 0.f;
      curGid = gid;
    }
#pragma unroll
    for (int nt = 0; nt < 4; ++nt)
      sum[nt] += fmaxf(acc[nt][rr] + sB[nt * 16 + r16], 0.0f);
  }
#pragma unroll
  for (int nt = 0; nt < 4; ++nt)
    atomicAdd(&out[(size_t)curGid * 64 + nt * 16 + r16], sum[nt]);
}

// ---------------------------------------------------------------------------
extern "C" void kernel_launch(void* const* d_in, const int* in_sizes, int n_in,
                              void* d_out, int out_size, void* d_ws,
                              size_t ws_size, hipStream_t stream) {
  const int* tokens    = (const int*)d_in[0];
  const int* edge_src  = (const int*)d_in[1];
  const int* edge_dst  = (const int*)d_in[2];
  const int* graph_ids = (const int*)d_in[3];
  const float* embed   = (const float*)d_in[4];
  const float* W1      = (const float*)d_in[5];
  const float* b1      = (const float*)d_in[6];
  const float* Wpool   = (const float*)d_in[7];
  const float* bpool   = (const float*)d_in[8];
  const float* Wself   = (const float*)d_in[9];
  const float* Wneigh  = (const float*)d_in[10];
  const float* bneigh  = (const float*)d_in[11];
  const float* Wlin    = (const float*)d_in[12];
  const float* blin    = (const float*)d_in[13];
  float* out = (float*)d_out;
  float* ws  = (float*)d_ws;

  const size_t N = N_NODES;
  // ws layout (floats). Zeroed region [0, 50N) holds the accumulated buffers.
  float* degO = ws;            // N   -> norm_src after norm_kernel
  float* degI = ws + N;        // N   -> norm_dst
  float* m    = ws + 2 * N;    // 16N (atomic-add target)
  float* hn   = ws + 18 * N;   // 32N (atomic-max target)
  float* x    = ws + 50 * N;   // 16N
  float* h1   = ws + 66 * N;   // 32N
  float* hp   = ws + 98 * N;   // 32N
  float* h2   = ws + 130 * N;  // 64N   (total 194N floats = 388 MB)

  const int T = 256;
  long z4 = (long)(50 * N) / 4;
  zero4_kernel<<<(int)((z4 + T - 1) / T), T, 0, stream>>>((float4*)ws, z4);
  long o4 = out_size / 4;
  zero4_kernel<<<(int)((o4 + T - 1) / T), T, 0, stream>>>((float4*)out, o4);

  degree_kernel<<<(N_EDGES + T - 1) / T, T, 0, stream>>>(edge_src, edge_dst,
                                                         degO, degI);
  norm_kernel<<<(N_NODES + T - 1) / T, T, 0, stream>>>(degO, degI);
  embed_scale_kernel<<<(int)(((long)N * 4 + T - 1) / T), T, 0, stream>>>(
      tokens, embed, degO, x);
  scatter_add_kernel<<<(int)(((long)N_EDGES * 4 + T - 1) / T), T, 0, stream>>>(
      edge_src, edge_dst, x, m);

  const int gemmBlocks = (N_TILES + 3) / 4;  // 4 waves (tiles) per 128-thr block
  gemm1_kernel<<<gemmBlocks, 128, 0, stream>>>(m, degI, W1, b1, h1);
  gemm_pool_kernel<<<gemmBlocks, 128, 0, stream>>>(h1, Wpool, bpool, hp);
  scatter_max_kernel<<<(int)(((long)N_EDGES * 8 + T - 1) / T), T, 0, stream>>>(
      edge_src, edge_dst, hp, hn);
  gemm2_kernel<<<gemmBlocks, 128, 0, stream>>>(h1, hn, Wself, Wneigh, bneigh,
                                               h2);
  gemm3_kernel<<<gemmBlocks, 128, 0, stream>>>(h2, Wlin, blin, graph_ids, out);
}